// LocalSpatialEncoding_12592844112371
// MI455X (gfx1250) — compile-verified
//
#include <hip/hip_runtime.h>

// Problem constants (match reference)
#define BB 4
#define NNpts 16384
#define KK 16
#define DD 16
#define CH 10
#define NBLK1 256          // blocks in stats pass
#define WPB 8              // wave32s per 256-thread block

typedef __attribute__((ext_vector_type(2))) float v2f;
typedef __attribute__((ext_vector_type(4))) float v4f;
typedef __attribute__((ext_vector_type(8))) float v8f;

struct WmmaA { v2f a0, a1, a2; };

// A-matrix (conv_w) per-lane registers, per ISA 32-bit A 16x4 layout:
// lanes 0-15: M=lane, VGPR0=K0, VGPR1=K1; lanes 16-31: M=lane-16, VGPR0=K2, VGPR1=K3.
__device__ __forceinline__ WmmaA load_A(const float* __restrict__ conv_w, int pm, int half) {
  const float* wrow = conv_w + pm * CH;
  WmmaA A;
  A.a0 = (v2f){ wrow[0 + 2 * half], wrow[1 + 2 * half] };           // chunk0: ch 0..3
  A.a1 = (v2f){ wrow[4 + 2 * half], wrow[5 + 2 * half] };           // chunk1: ch 4..7
  A.a2 = half ? (v2f){0.f, 0.f} : (v2f){ wrow[8], wrow[9] };        // chunk2: ch 8,9,pad,pad
  return A;
}

// One tile = one (b,n): 16 neighbors (k = pm), center coords shared.
// Builds B (4x16 per chunk, mirrored layout) and chains 3 f32 WMMAs.
__device__ __forceinline__ v8f tile_conv(const float* __restrict__ coords,
                                         const int* __restrict__ idx,
                                         const float* __restrict__ dist,
                                         const WmmaA& A, v8f c,
                                         int b, int n, int pm, int half) {
  const float* cc = coords + (size_t)(b * NNpts + n) * 3;
  float cx = cc[0], cy = cc[1], cz = cc[2];
  int j = idx[((size_t)b * NNpts + n) * KK + pm];
  const float* nc = coords + (size_t)(b * NNpts + j) * 3;
  float nx0 = nc[0], ny0 = nc[1], nz0 = nc[2];
  float dd = dist[((size_t)b * NNpts + n) * KK + pm];
  // lo half feeds K0/K1 (ch 0,1 | 4,5 | 8,9), hi half feeds K2/K3 (ch 2,3 | 6,7 | pad)
  v2f b0 = half ? (v2f){cz, nx0}              : (v2f){cx, cy};
  v2f b1 = half ? (v2f){cx - nx0, cy - ny0}   : (v2f){ny0, nz0};
  v2f b2 = half ? (v2f){0.f, 0.f}             : (v2f){cz - nz0, dd};
  c = __builtin_amdgcn_wmma_f32_16x16x4_f32(false, A.a0, false, b0, (short)0, c, false, false);
  c = __builtin_amdgcn_wmma_f32_16x16x4_f32(false, A.a1, false, b1, (short)0, c, false, false);
  c = __builtin_amdgcn_wmma_f32_16x16x4_f32(false, A.a2, false, b2, (short)0, c, false, false);
  return c;
}

// Pass 1: compute conv output on the fly, accumulate per-channel sum/sumsq.
// Deterministic: register accumulation -> fixed butterfly within 16-lane halves
// -> LDS -> per-block partials written to ws[32 + blk*32 + {0..31}].
__global__ __launch_bounds__(256) void lse_stats_kernel(
    const float* __restrict__ coords, const int* __restrict__ idx,
    const float* __restrict__ dist, const float* __restrict__ conv_w,
    const float* __restrict__ conv_b, float* __restrict__ ws) {
  __shared__ float lds[WPB * 64];
  const int lane = threadIdx.x & 31;
  const int wave = threadIdx.x >> 5;
  const int pm = lane & 15, half = lane >> 4;

  WmmaA A = load_A(conv_w, pm, half);
  v8f cinit;
#pragma unroll
  for (int r = 0; r < 8; ++r) cinit[r] = conv_b[r + 8 * half];

  float acc[8], accsq[8];
#pragma unroll
  for (int r = 0; r < 8; ++r) { acc[r] = 0.f; accsq[r] = 0.f; }

  const int gwave = blockIdx.x * WPB + wave;
  const int nwaves = NBLK1 * WPB;            // 2048
  const int ntiles = BB * NNpts;             // 65536 -> exactly 32 tiles/wave (uniform)
  for (int t = gwave; t < ntiles; t += nwaves) {
    int b = t >> 14;                         // t / NNpts
    int n = t & (NNpts - 1);
    v8f c = tile_conv(coords, idx, dist, A, cinit, b, n, pm, half);
#pragma unroll
    for (int r = 0; r < 8; ++r) { acc[r] += c[r]; accsq[r] += c[r] * c[r]; }
  }

  // reduce over the 16 points within each half-wave (channels differ between halves)
#pragma unroll
  for (int r = 0; r < 8; ++r) {
#pragma unroll
    for (int m = 1; m < 16; m <<= 1) {
      acc[r]   += __shfl_xor(acc[r],   m, 16);
      accsq[r] += __shfl_xor(accsq[r], m, 16);
    }
  }
  if (pm == 0) {
#pragma unroll
    for (int r = 0; r < 8; ++r) {
      int d = r + 8 * half;
      lds[wave * 64 + d]      = acc[r];
      lds[wave * 64 + 32 + d] = accsq[r];
    }
  }
  __syncthreads();
  if (threadIdx.x < 32) {
    int d = threadIdx.x & 15;
    int isq = threadIdx.x >> 4;
    float s = 0.f;
#pragma unroll
    for (int w = 0; w < WPB; ++w) s += lds[w * 64 + isq * 32 + d];
    ws[32 + blockIdx.x * 32 + isq * 16 + d] = s;
  }
}

// Pass 2: fold block partials (fixed order) into per-channel scale/shift.
__global__ void lse_finalize_stats(const float* __restrict__ bn_gamma,
                                   const float* __restrict__ bn_beta,
                                   float* __restrict__ ws) {
  int d = threadIdx.x;
  if (d >= DD) return;
  float s = 0.f, sq = 0.f;
  for (int blk = 0; blk < NBLK1; ++blk) {
    s  += ws[32 + blk * 32 + d];
    sq += ws[32 + blk * 32 + 16 + d];
  }
  const float M = (float)(BB * NNpts * KK);
  float mean = s / M;
  float var = sq / M - mean * mean;
  float rs = rsqrtf(var + 1e-6f);
  float scale = bn_gamma[d] * rs;
  ws[d] = scale;
  ws[16 + d] = bn_beta[d] - mean * scale;
}

// Pass 3: recompute conv via WMMA, apply BN scale/shift + ReLU, store channels
// 0..15; broadcast `features` into channels 16..31. One tile per wave.
__global__ __launch_bounds__(256) void lse_output_kernel(
    const float* __restrict__ coords, const float* __restrict__ features,
    const int* __restrict__ idx, const float* __restrict__ dist,
    const float* __restrict__ conv_w, const float* __restrict__ conv_b,
    const float* __restrict__ ws, float* __restrict__ out) {
  const int lane = threadIdx.x & 31;
  const int wave = threadIdx.x >> 5;
  const int pm = lane & 15, half = lane >> 4;

  WmmaA A = load_A(conv_w, pm, half);
  v8f cinit;
  float sc[8], sh[8];
#pragma unroll
  for (int r = 0; r < 8; ++r) {
    int d = r + 8 * half;
    cinit[r] = conv_b[d];
    sc[r] = ws[d];
    sh[r] = ws[16 + d];
  }

  const int t = blockIdx.x * WPB + wave;     // tile id, exactly B*N waves launched
  const int b = t >> 14;
  const int n = t & (NNpts - 1);

  v8f c = tile_conv(coords, idx, dist, A, cinit, b, n, pm, half);

  const size_t outbase = (((size_t)b * 2 * DD) * NNpts + n) * KK;
#pragma unroll
  for (int r = 0; r < 8; ++r) {
    int d = r + 8 * half;                    // D row r, lanes hi are rows 8..15
    float v = c[r] * sc[r] + sh[r];
    v = v > 0.f ? v : 0.f;
    out[outbase + (size_t)d * (NNpts * KK) + pm] = v;
  }
  // features half: channel 16+pm gets features[b, pm, n] broadcast over k;
  // lo half writes k=0..7, hi half k=8..15 (two 16B stores of the splat).
  float f = features[((size_t)b * DD + pm) * NNpts + n];
  size_t fbase = outbase + (size_t)(DD + pm) * (NNpts * KK) + half * 8;
  *(v4f*)(out + fbase)     = (v4f){f, f, f, f};
  *(v4f*)(out + fbase + 4) = (v4f){f, f, f, f};
}

extern "C" void kernel_launch(void* const* d_in, const int* in_sizes, int n_in,
                              void* d_out, int out_size, void* d_ws, size_t ws_size,
                              hipStream_t stream) {
  const float* coords   = (const float*)d_in[0];
  const float* features = (const float*)d_in[1];
  const int*   idx      = (const int*)  d_in[2];
  const float* dist     = (const float*)d_in[3];
  const float* conv_w   = (const float*)d_in[4];
  const float* conv_b   = (const float*)d_in[5];
  const float* bn_gamma = (const float*)d_in[6];
  const float* bn_beta  = (const float*)d_in[7];
  float* out = (float*)d_out;
  float* ws  = (float*)d_ws;   // [0..15]=scale [16..31]=shift [32..32+256*32)=block partials

  lse_stats_kernel<<<NBLK1, 256, 0, stream>>>(coords, idx, dist, conv_w, conv_b, ws);
  lse_finalize_stats<<<1, 16, 0, stream>>>(bn_gamma, bn_beta, ws);
  lse_output_kernel<<<(BB * NNpts) / WPB, 256, 0, stream>>>(
      coords, features, idx, dist, conv_w, conv_b, ws, out);
}